// GrayDocument_2688649528013
// MI455X (gfx1250) — compile-verified
//
#include <hip/hip_runtime.h>
#include <stdint.h>

// ---------------------------------------------------------------------------
// GrayDocument pipeline for MI455X (gfx1250):
//   K1: TDM-streamed channel-mean + per-block min/max/sum partials
//   Kf: per-image affine finalize (min/max/autoinvert folded into A,B)
//   K2: TDM-streamed apply: out = clip(A*gray + B + 0.01*gauss(idx))
// Memory-bound op (AI ~ 10 flops / 16B): the CDNA5 win is tensor_load_to_lds
// DMA double-buffering (TENSORcnt pipelining) + wave32 shuffle reductions.
// ---------------------------------------------------------------------------

typedef float    v4f __attribute__((ext_vector_type(4)));
typedef uint32_t v4u __attribute__((ext_vector_type(4)));
typedef int32_t  v8i __attribute__((ext_vector_type(8)));
typedef int32_t  v4i __attribute__((ext_vector_type(4)));

constexpr int kImages         = 64;
constexpr int kPlane          = 768 * 768;            // 589824 pixels / channel
constexpr int kTile           = 2048;                 // pixels per TDM tile
constexpr int kChunksPerBlock = 8;                    // 16384 pixels per block
constexpr int kBlocksPerImage = kPlane / (kTile * kChunksPerBlock); // 36
constexpr int kGrid           = kImages * kBlocksPerImage;          // 2304
constexpr int kThreads        = 256;                  // 8 wave32s
constexpr float kNoise        = 0.01f;

// ---- Tensor Data Mover: load a 3-channel x kTile-pixel tile into LDS -------
// D# per CDNA5 ISA:
//   group0: count=1 | lds_addr | global_addr(57b) | type=2
//   group1: data_size=4B, tensor_dim0=kPlane, tensor_dim1=3,
//           tile_dim0=kTile, tile_dim1=3, tensor_dim0_stride=kPlane
// One DMA therefore gathers the three channel slices of a chunk into
// contiguous LDS (row y at lds_off + y*kTile*4).
__device__ __forceinline__ void tdm_load_tile(const float* gptr, uint32_t lds_off) {
  uint64_t ga = (uint64_t)(uintptr_t)gptr;
  v4u g0;
  g0[0] = 1u;                                              // count=1, user mode
  g0[1] = lds_off;                                         // LDS byte address
  g0[2] = (uint32_t)ga;                                    // global_addr[31:0]
  g0[3] = ((uint32_t)(ga >> 32) & 0x01FFFFFFu) | (2u << 30); // addr[56:32] | type=2
  v8i g1;
  g1[0] = (int32_t)(2u << 16);                             // data_size = 4 bytes
  g1[1] = (int32_t)(((uint32_t)kPlane & 0xFFFFu) << 16);   // tensor_dim0[15:0]
  g1[2] = (int32_t)((((uint32_t)kPlane >> 16) & 0xFFFFu)   // tensor_dim0[31:16]
                    | (3u << 16));                         // tensor_dim1 = 3
  g1[3] = (int32_t)((uint32_t)kTile << 16);                // tile_dim0 = kTile
  g1[4] = 3;                                               // tile_dim1=3, tile_dim2=0
  g1[5] = kPlane;                                          // dim0_stride[31:0]
  g1[6] = 0;                                               // dim0_stride[47:32], dim1_stride lo
  g1[7] = 0;
  v4i g2 = {0, 0, 0, 0};
  v4i g3 = {0, 0, 0, 0};
#if defined(__clang_major__) && (__clang_major__ >= 23)
  v8i g4 = {0, 0, 0, 0, 0, 0, 0, 0};
  __builtin_amdgcn_tensor_load_to_lds(g0, g1, g2, g3, g4, 0);
#else
  __builtin_amdgcn_tensor_load_to_lds(g0, g1, g2, g3, 0);
#endif
}

// ---- deterministic per-index Gaussian (PCG hash + Box-Muller) --------------
__device__ __forceinline__ uint32_t pcg(uint32_t v) {
  uint32_t s = v * 747796405u + 2891336453u;
  uint32_t w = ((s >> ((s >> 28u) + 4u)) ^ s) * 277803737u;
  return (w >> 22u) ^ w;
}

__device__ __forceinline__ float gauss01(uint32_t idx) {
  uint32_t h1 = pcg(idx ^ 0x68bc21ebu);
  uint32_t h2 = pcg(h1 + 0x9e3779b9u);
  float u1 = (float)((h1 >> 8) + 1u) * (1.0f / 16777216.0f);  // (0,1]
  float u2 = (float)(h2 >> 8) * (1.0f / 16777216.0f);         // [0,1)
  float r  = __builtin_sqrtf(-2.0f * __logf(u1));
  return r * __cosf(6.2831853071795864f * u2);
}

// ---- K1: gray stats (min / max / sum) per block ----------------------------
__global__ void __launch_bounds__(kThreads)
k_stats(const float* __restrict__ a,
        float* __restrict__ pmin, float* __restrict__ pmax, float* __restrict__ psum) {
  __shared__ __align__(16) float tile[2][3 * kTile];   // 48 KB double buffer
  __shared__ float sMin[kThreads / 32], sMax[kThreads / 32], sSum[kThreads / 32];

  const int tid = threadIdx.x;
  const int img = blockIdx.x / kBlocksPerImage;
  const int blk = blockIdx.x % kBlocksPerImage;
  const float* base = a + (size_t)img * (3 * (size_t)kPlane)
                        + (size_t)blk * (kChunksPerBlock * kTile);

  float vmin = __builtin_inff(), vmax = -__builtin_inff(), vsum = 0.0f;

  for (int c = 0; c < kChunksPerBlock; ++c) {
    if (tid < 32) {                      // wave 0 drives the DMA pipeline
      if (c == 0)
        tdm_load_tile(base, (uint32_t)(uintptr_t)&tile[0][0]);
      if (c + 1 < kChunksPerBlock) {
        tdm_load_tile(base + (size_t)(c + 1) * kTile,
                      (uint32_t)(uintptr_t)&tile[(c + 1) & 1][0]);
        __builtin_amdgcn_s_wait_tensorcnt((short)1);   // chunk c landed, c+1 in flight
      } else {
        __builtin_amdgcn_s_wait_tensorcnt((short)0);
      }
    }
    __syncthreads();

    const v4f* c0 = (const v4f*)&tile[c & 1][0];
    const v4f* c1 = (const v4f*)&tile[c & 1][kTile];
    const v4f* c2 = (const v4f*)&tile[c & 1][2 * kTile];
#pragma unroll
    for (int j = 0; j < 2; ++j) {
      int q = tid * 2 + j;                               // 8 consecutive pixels/thread
      v4f g = (c0[q] + c1[q] + c2[q]) * (1.0f / 3.0f);
      vsum += g[0] + g[1] + g[2] + g[3];
      vmin = fminf(vmin, fminf(fminf(g[0], g[1]), fminf(g[2], g[3])));
      vmax = fmaxf(vmax, fmaxf(fmaxf(g[0], g[1]), fmaxf(g[2], g[3])));
    }
    __syncthreads();                                     // buffer reuse fence
  }

  // wave32 butterfly reduction
#pragma unroll
  for (int off = 16; off > 0; off >>= 1) {
    vsum += __shfl_xor(vsum, off, 32);
    vmin = fminf(vmin, __shfl_xor(vmin, off, 32));
    vmax = fmaxf(vmax, __shfl_xor(vmax, off, 32));
  }
  const int wave = tid >> 5, lane = tid & 31;
  if (lane == 0) { sMin[wave] = vmin; sMax[wave] = vmax; sSum[wave] = vsum; }
  __syncthreads();
  if (tid == 0) {
    float m = sMin[0], M = sMax[0], S = sSum[0];
#pragma unroll
    for (int w = 1; w < kThreads / 32; ++w) {
      m = fminf(m, sMin[w]); M = fmaxf(M, sMax[w]); S += sSum[w];
    }
    pmin[blockIdx.x] = m; pmax[blockIdx.x] = M; psum[blockIdx.x] = S;
  }
}

// ---- Kf: fold 36 partials per image into affine (A, B) ---------------------
__global__ void k_finalize(const float* __restrict__ pmin, const float* __restrict__ pmax,
                           const float* __restrict__ psum, float* __restrict__ ab) {
  const int img = threadIdx.x;           // 1 block x 64 threads
  if (img >= kImages) return;
  float mn = __builtin_inff(), mxv = -__builtin_inff(), s = 0.0f;
  for (int i = 0; i < kBlocksPerImage; ++i) {
    const int k = img * kBlocksPerImage + i;
    mn = fminf(mn, pmin[k]); mxv = fmaxf(mxv, pmax[k]); s += psum[k];
  }
  const float mx   = fmaxf(0.5f, mxv - mn);
  const float mean = (s * (1.0f / (float)kPlane) - mn) / mx;
  float A, B;
  if (mean > 0.5f) { A = -1.0f / mx; B = 1.0f + mn / mx; }   // inverted
  else             { A =  1.0f / mx; B = -mn / mx; }
  ab[2 * img]     = A;
  ab[2 * img + 1] = B;
}

// ---- K2: apply affine + noise + clip, same TDM pipeline --------------------
__global__ void __launch_bounds__(kThreads)
k_apply(const float* __restrict__ a, const float* __restrict__ ab,
        float* __restrict__ out) {
  __shared__ __align__(16) float tile[2][3 * kTile];

  const int tid = threadIdx.x;
  const int img = blockIdx.x / kBlocksPerImage;
  const int blk = blockIdx.x % kBlocksPerImage;
  const float* base = a + (size_t)img * (3 * (size_t)kPlane)
                        + (size_t)blk * (kChunksPerBlock * kTile);
  const float A = ab[2 * img];
  const float B = ab[2 * img + 1];
  const size_t outBase = (size_t)img * kPlane + (size_t)blk * (kChunksPerBlock * kTile);

  for (int c = 0; c < kChunksPerBlock; ++c) {
    if (tid < 32) {
      if (c == 0)
        tdm_load_tile(base, (uint32_t)(uintptr_t)&tile[0][0]);
      if (c + 1 < kChunksPerBlock) {
        tdm_load_tile(base + (size_t)(c + 1) * kTile,
                      (uint32_t)(uintptr_t)&tile[(c + 1) & 1][0]);
        __builtin_amdgcn_s_wait_tensorcnt((short)1);
      } else {
        __builtin_amdgcn_s_wait_tensorcnt((short)0);
      }
    }
    __syncthreads();

    const v4f* c0 = (const v4f*)&tile[c & 1][0];
    const v4f* c1 = (const v4f*)&tile[c & 1][kTile];
    const v4f* c2 = (const v4f*)&tile[c & 1][2 * kTile];
    v4f* outv = (v4f*)(out + outBase + (size_t)c * kTile);
#pragma unroll
    for (int j = 0; j < 2; ++j) {
      int q = tid * 2 + j;
      v4f g = (c0[q] + c1[q] + c2[q]) * (1.0f / 3.0f);
      v4f v = g * A + B;
      const uint32_t idx0 = (uint32_t)(outBase + (size_t)c * kTile) + (uint32_t)q * 4u;
#pragma unroll
      for (int k = 0; k < 4; ++k) {
        float x = v[k] + kNoise * gauss01(idx0 + (uint32_t)k);
        v[k] = fminf(fmaxf(x, 0.0f), 1.0f);
      }
      outv[q] = v;
    }
    __syncthreads();
  }
}

// ---------------------------------------------------------------------------
extern "C" void kernel_launch(void* const* d_in, const int* in_sizes, int n_in,
                              void* d_out, int out_size, void* d_ws, size_t ws_size,
                              hipStream_t stream) {
  (void)in_sizes; (void)n_in; (void)out_size; (void)ws_size;
  const float* a = (const float*)d_in[0];
  float* out = (float*)d_out;

  float* pmin = (float*)d_ws;            // [kGrid]
  float* pmax = pmin + kGrid;            // [kGrid]
  float* psum = pmax + kGrid;            // [kGrid]
  float* ab   = psum + kGrid;            // [2 * kImages]

  k_stats<<<kGrid, kThreads, 0, stream>>>(a, pmin, pmax, psum);
  k_finalize<<<1, 64, 0, stream>>>(pmin, pmax, psum, ab);
  k_apply<<<kGrid, kThreads, 0, stream>>>(a, ab, out);
}